// NodeClassifier_80144089743763
// MI455X (gfx1250) — compile-verified
//
#include <hip/hip_runtime.h>
#include <hip/hip_bf16.h>

#define N_NODES   50000
#define N_EDGES   800000
#define HID_DIM   128
#define N_CLASSES 32
#define BN_EPS    1e-5f

typedef __attribute__((ext_vector_type(2))) float v2f;
typedef __attribute__((ext_vector_type(8))) float v8f;

// ---------------------------------------------------------------------------
// float4 copy: out = in  (n = count of float4)
// ---------------------------------------------------------------------------
__global__ void copy4_kernel(const float4* __restrict__ in, float4* __restrict__ out, int n) {
    int i = blockIdx.x * blockDim.x + threadIdx.x;
    if (i < n) out[i] = in[i];
}

// ---------------------------------------------------------------------------
// Edge scatter-add, 128 features: out[dst[e]][:] += h[src[e]][:]
// One wave (32 lanes) per edge; each lane moves a float4 (32*4 = 128 feats).
// unsafeAtomicAdd -> GLOBAL_ATOMIC_ADD_F32 (hardware fp32 atomic at L2).
// ---------------------------------------------------------------------------
__global__ void scatter_add_kernel(const float* __restrict__ h,
                                   const int* __restrict__ ei,
                                   float* __restrict__ out) {
    int t    = blockIdx.x * blockDim.x + threadIdx.x;
    int e    = t >> 5;
    int lane = t & 31;
    int src = ei[e];
    int dst = ei[N_EDGES + e];
    const float4 v = *(const float4*)(h + (size_t)src * HID_DIM + lane * 4);
    float* o = out + (size_t)dst * HID_DIM + lane * 4;
    unsafeAtomicAdd(o + 0, v.x);
    unsafeAtomicAdd(o + 1, v.y);
    unsafeAtomicAdd(o + 2, v.z);
    unsafeAtomicAdd(o + 3, v.w);
}

// ---------------------------------------------------------------------------
// Edge scatter-add, 32 features (post-projection): 8 lanes per edge.
// agg(h) @ W2 == agg(h @ W2)  ->  4x fewer atomics than aggregating at 128-d.
// ---------------------------------------------------------------------------
__global__ void scatter_add32_kernel(const float* __restrict__ g,
                                     const int* __restrict__ ei,
                                     float* __restrict__ out) {
    int t    = blockIdx.x * blockDim.x + threadIdx.x;
    int e    = t >> 3;
    int lane = t & 7;
    int src = ei[e];
    int dst = ei[N_EDGES + e];
    const float4 v = *(const float4*)(g + (size_t)src * N_CLASSES + lane * 4);
    float* o = out + (size_t)dst * N_CLASSES + lane * 4;
    unsafeAtomicAdd(o + 0, v.x);
    unsafeAtomicAdd(o + 1, v.y);
    unsafeAtomicAdd(o + 2, v.z);
    unsafeAtomicAdd(o + 3, v.w);
}

// ---------------------------------------------------------------------------
// GEMM1: out[50000x128] = H[50000x128] @ W[128x128] + bias
// 256 threads = 8 waves; block computes a 16x128 row panel; wave w computes
// the 16x16 tile at columns [16w,16w+16) with V_WMMA_F32_16X16X4_F32.
// Lane layouts (ISA 7.12.2, 32-bit):
//  A 16x4 : lanes 0-15 M=0..15 {K=k,k+1}; lanes 16-31 M=0..15 {K=k+2,k+3}
//  B 4x16 : v0: lanes0-15 row k  / lanes16-31 row k+2
//           v1: lanes0-15 row k+1/ lanes16-31 row k+3
//  C 16x16: vgpr v: lanes0-15 -> M=v, lanes16-31 -> M=v+8 (N = lane&15)
// ---------------------------------------------------------------------------
__global__ void gemm1_kernel(const float* __restrict__ H,
                             const float* __restrict__ W,
                             const float* __restrict__ bias,
                             float* __restrict__ out) {
    __shared__ float sA[16 * 128];   // 8 KB
    __shared__ float sW[64 * 128];   // 32 KB (K-chunk of W)

    const int tid     = threadIdx.x;
    const int rowBase = blockIdx.x * 16;

    {   // Stage 16x128 A panel (contiguous 2048 floats)
        const float4* srcA = (const float4*)(H + (size_t)rowBase * 128);
        float4* dA = (float4*)sA;
        for (int i = tid; i < 512; i += 256) dA[i] = srcA[i];
    }

    const int wave = tid >> 5;
    const int lane = tid & 31;
    const int half = lane >> 4;
    const int mn   = lane & 15;
    const int n0   = wave * 16;

    v8f c = {};
    for (int kc = 0; kc < 128; kc += 64) {
        __syncthreads();                       // prior chunk consumed / sA visible
        const float4* srcW = (const float4*)(W + (size_t)kc * 128);
        float4* dW = (float4*)sW;
        for (int i = tid; i < 2048; i += 256) dW[i] = srcW[i];
        __syncthreads();

        for (int k = 0; k < 64; k += 4) {
            v2f a, b;
            a[0] = sA[mn * 128 + kc + k + 2 * half];
            a[1] = sA[mn * 128 + kc + k + 2 * half + 1];
            b[0] = sW[(k + 2 * half) * 128 + n0 + mn];
            b[1] = sW[(k + 2 * half + 1) * 128 + n0 + mn];
            c = __builtin_amdgcn_wmma_f32_16x16x4_f32(false, a, false, b,
                                                      (short)0, c, false, false);
        }
    }

    const int col = n0 + mn;
    const float bv = bias[col];
    #pragma unroll
    for (int v = 0; v < 8; v++) {
        int row = rowBase + v + 8 * half;
        out[(size_t)row * 128 + col] = c[v] + bv;
    }
}

// ---------------------------------------------------------------------------
// GEMM2 (no bias): out[50000x32] = H[50000x128] @ W[128x32]
// Bias is applied after aggregation (linearity), inside the softmax kernel.
// One block = 64 threads = 2 waves; block computes a 16x32 row panel.
// ---------------------------------------------------------------------------
__global__ void gemm2_kernel(const float* __restrict__ H,
                             const float* __restrict__ W,
                             float* __restrict__ out) {
    __shared__ float sA[16 * 128];   // 8 KB
    __shared__ float sW[128 * 32];   // 16 KB (all of W2)

    const int tid     = threadIdx.x;   // 0..63
    const int rowBase = blockIdx.x * 16;

    {
        const float4* srcA = (const float4*)(H + (size_t)rowBase * 128);
        float4* dA = (float4*)sA;
        for (int i = tid; i < 512; i += 64) dA[i] = srcA[i];
        const float4* srcW = (const float4*)W;
        float4* dW = (float4*)sW;
        for (int i = tid; i < 1024; i += 64) dW[i] = srcW[i];
    }
    __syncthreads();

    const int wave = tid >> 5;
    const int lane = tid & 31;
    const int half = lane >> 4;
    const int mn   = lane & 15;
    const int n0   = wave * 16;

    v8f c = {};
    for (int k = 0; k < 128; k += 4) {
        v2f a, b;
        a[0] = sA[mn * 128 + k + 2 * half];
        a[1] = sA[mn * 128 + k + 2 * half + 1];
        b[0] = sW[(k + 2 * half) * 32 + n0 + mn];
        b[1] = sW[(k + 2 * half + 1) * 32 + n0 + mn];
        c = __builtin_amdgcn_wmma_f32_16x16x4_f32(false, a, false, b,
                                                  (short)0, c, false, false);
    }

    const int col = n0 + mn;
    #pragma unroll
    for (int v = 0; v < 8; v++) {
        int row = rowBase + v + 8 * half;
        out[(size_t)row * 32 + col] = c[v];
    }
}

// ---------------------------------------------------------------------------
// BatchNorm stats pass: stats[f] += sum_n h[n][f], stats[128+f] += sum h^2.
// 256 threads = 2 node-groups x 128 features; LDS pair-reduce, then fp32
// global atomics (stats zeroed by hipMemsetAsync before launch).
// ---------------------------------------------------------------------------
__global__ void bn_stats_kernel(const float* __restrict__ h, float* __restrict__ stats) {
    __shared__ float sSum[256];
    __shared__ float sSq[256];
    const int tid = threadIdx.x;
    const int f   = tid & 127;
    const int g   = tid >> 7;

    float s = 0.f, q = 0.f;
    for (int node = blockIdx.x * 2 + g; node < N_NODES; node += gridDim.x * 2) {
        float v = h[(size_t)node * 128 + f];
        s += v;
        q += v * v;
    }
    sSum[tid] = s;
    sSq[tid]  = q;
    __syncthreads();
    if (tid < 128) {
        unsafeAtomicAdd(&stats[f],       sSum[tid] + sSum[tid + 128]);
        unsafeAtomicAdd(&stats[128 + f], sSq[tid]  + sSq[tid + 128]);
    }
}

// stats[256+f] = scale, stats[384+f] = shift  (fused BN affine form)
__global__ void bn_finalize_kernel(float* __restrict__ stats,
                                   const float* __restrict__ gamma,
                                   const float* __restrict__ beta) {
    int f = threadIdx.x;
    float mean  = stats[f] * (1.0f / N_NODES);
    float var   = stats[128 + f] * (1.0f / N_NODES) - mean * mean;  // biased var
    float scale = gamma[f] * rsqrtf(var + BN_EPS);
    stats[256 + f] = scale;
    stats[384 + f] = beta[f] - mean * scale;
}

// out = selu(h * scale[f] + shift[f]), elementwise over 50000*128
__global__ void bn_selu_kernel(const float* __restrict__ h,
                               const float* __restrict__ stats,
                               float* __restrict__ out) {
    int i = blockIdx.x * blockDim.x + threadIdx.x;
    int f = i & 127;
    float v = h[i] * stats[256 + f] + stats[384 + f];
    const float lam   = 1.0507009873554805f;
    const float alpha = 1.6732632423543772f;
    out[i] = v > 0.f ? lam * v : lam * alpha * (expf(v) - 1.0f);
}

// ---------------------------------------------------------------------------
// Row softmax over 32 classes with deferred bias: one wave32 per node.
// ---------------------------------------------------------------------------
__global__ void softmax32_kernel(const float* __restrict__ logits,
                                 const float* __restrict__ bias,
                                 float* __restrict__ out) {
    int node = blockIdx.x * 8 + (threadIdx.x >> 5);
    int lane = threadIdx.x & 31;
    float v = logits[(size_t)node * 32 + lane] + bias[lane];
    float m = v;
    #pragma unroll
    for (int o = 16; o > 0; o >>= 1) m = fmaxf(m, __shfl_xor(m, o, 32));
    float e = expf(v - m);
    float s = e;
    #pragma unroll
    for (int o = 16; o > 0; o >>= 1) s += __shfl_xor(s, o, 32);
    out[(size_t)node * 32 + lane] = e / s;
}

// ---------------------------------------------------------------------------
extern "C" void kernel_launch(void* const* d_in, const int* in_sizes, int n_in,
                              void* d_out, int out_size, void* d_ws, size_t ws_size,
                              hipStream_t stream) {
    const float* x     = (const float*)d_in[0];
    const int*   ei    = (const int*)  d_in[1];   // [2, N_EDGES] int32
    const float* W1    = (const float*)d_in[2];
    const float* b1    = (const float*)d_in[3];
    const float* gamma = (const float*)d_in[4];
    const float* beta  = (const float*)d_in[5];
    const float* W2    = (const float*)d_in[6];
    const float* b2    = (const float*)d_in[7];
    float* out = (float*)d_out;

    const size_t NF = (size_t)N_NODES * HID_DIM;       // 6.4M floats
    const size_t NC = (size_t)N_NODES * N_CLASSES;     // 1.6M floats
    float* A     = (float*)d_ws;                       // node x 128
    float* B     = A + NF;                             // node x 128
    float* C     = B + NF;                             // node x 128
    float* stats = C + NF;                             // 512 floats
    float* G     = stats + 512;                        // node x 32  (projection)
    float* L     = G + NC;                             // node x 32  (agg'd logits)

    const int nF4       = (int)(NF / 4);               // 1.6M float4
    const int nC4       = (int)(NC / 4);               // 400K float4
    const int copyGrid  = nF4 / 256;                   // 6250 (exact)
    const int copyGrid2 = (nC4 + 255) / 256;           // 1563
    const int scatGrid  = (N_EDGES * 32) / 256;        // 100000
    const int scatGrid2 = (N_EDGES * 8) / 256;         // 25000
    const int mTiles    = N_NODES / 16;                // 3125 (exact)

    // agg #1: A = x + scatter(x)          (128-d)
    copy4_kernel<<<copyGrid, 256, 0, stream>>>((const float4*)x, (float4*)A, nF4);
    scatter_add_kernel<<<scatGrid, 256, 0, stream>>>(x, ei, A);

    // agg #2: B = A + scatter(A)          (128-d)
    copy4_kernel<<<copyGrid, 256, 0, stream>>>((const float4*)A, (float4*)B, nF4);
    scatter_add_kernel<<<scatGrid, 256, 0, stream>>>(A, ei, B);

    // linear1: C = B @ W1 + b1            (WMMA f32)
    gemm1_kernel<<<mTiles, 256, 0, stream>>>(B, W1, b1, C);

    // batchnorm stats + fused scale/shift
    hipMemsetAsync(stats, 0, 512 * sizeof(float), stream);
    bn_stats_kernel<<<500, 256, 0, stream>>>(C, stats);
    bn_finalize_kernel<<<1, 128, 0, stream>>>(stats, gamma, beta);

    // A = selu(bn(C))
    bn_selu_kernel<<<(int)(NF / 256), 256, 0, stream>>>(C, stats, A);

    // project FIRST: G = A @ W2           (WMMA f32; bias deferred)
    gemm2_kernel<<<mTiles, 64, 0, stream>>>(A, W2, G);

    // agg #3 at 32-d: L = G + scatter(G)  (4x fewer atomics than 128-d)
    copy4_kernel<<<copyGrid2, 256, 0, stream>>>((const float4*)G, (float4*)L, nC4);
    scatter_add32_kernel<<<scatGrid2, 256, 0, stream>>>(G, ei, L);

    // softmax(L + b2) -> d_out
    softmax32_kernel<<<N_NODES / 8, 256, 0, stream>>>(L, b2, out);
}